// HeatmapConsistencyLoss_34789235097872
// MI455X (gfx1250) — compile-verified
//
#include <hip/hip_runtime.h>
#include <math.h>

#define H_LR 48
#define W_LR 48
#define H_HR 128
#define W_HR 128
#define N_LR (H_LR * W_LR)      /* 2304  */
#define N_HR (H_HR * W_HR)      /* 16384 */
#define LOG_2PI 1.8378770664093453f
#define NSLICE 16
#define JT_PER_SLICE ((N_HR / 16) / NSLICE)   /* 64 j-tiles of 16 cols per slice */

typedef float v2f __attribute__((ext_vector_type(2)));
typedef float v8f __attribute__((ext_vector_type(8)));

// Single-exp streaming logsumexp update: state (m,s), new value x.
// Exactly one of exp(m-nm), exp(x-nm) is nontrivial -> one v_exp_f32.
__device__ __forceinline__ void lse_push(float& m, float& s, float x) {
    float nm = fmaxf(m, x);
    float lo = fminf(m, x);
    float e  = __expf(lo - nm);
    s = (x > m) ? fmaf(s, e, 1.0f) : (s + e);
    m = nm;
}

// Merge two (m,s) partials with one exp.
__device__ __forceinline__ void lse_merge(float& m, float& s, float om, float os) {
    float nm = fmaxf(m, om);
    float lo = fminf(m, om);
    float e  = __expf(lo - nm);
    s = (om > m) ? fmaf(s, e, os) : fmaf(os, e, s);
    m = nm;
}

// ---------------------------------------------------------------------------
// Prep HR: faithful log_softmax(softmax(pi_hr)) and per-component GEMM params.
// Record j (8 floats, 32B): [ -0.5/s0^2, m0/s0^2, -0.5/s1^2, m1/s1^2, D, 0,0,0 ]
// D = -0.5*(m0^2/s0^2 + m1^2/s1^2) - log s0 - log s1 - LOG_2PI + log_mix(j)
// ---------------------------------------------------------------------------
__global__ __launch_bounds__(1024)
void prep_hr_kernel(const float* __restrict__ mu_hr, const float* __restrict__ sig_hr,
                    const float* __restrict__ pi_hr, const float* __restrict__ loc_hr,
                    float* __restrict__ hrp) {
    __shared__ float red[1024];
    __shared__ float s_maxpi, s_sumE, s_logZ;
    const int t = threadIdx.x;

    // 1) max(pi_hr)
    float mx = -3.0e38f;
    for (int j = t; j < N_HR; j += 1024) mx = fmaxf(mx, pi_hr[j]);
    red[t] = mx; __syncthreads();
    for (int off = 512; off > 0; off >>= 1) {
        if (t < off) red[t] = fmaxf(red[t], red[t + off]);
        __syncthreads();
    }
    if (t == 0) s_maxpi = red[0];
    __syncthreads();
    const float maxpi = s_maxpi;

    // 2) softmax denominator
    float se = 0.f;
    for (int j = t; j < N_HR; j += 1024) se += __expf(pi_hr[j] - maxpi);
    red[t] = se; __syncthreads();
    for (int off = 512; off > 0; off >>= 1) {
        if (t < off) red[t] += red[t + off];
        __syncthreads();
    }
    if (t == 0) s_sumE = red[0];
    __syncthreads();
    const float invSum = 1.0f / s_sumE;

    // 3) Z = sum_j exp(p_j), p in (0,1] so no overflow; log_mix = p - log Z
    float z = 0.f;
    for (int j = t; j < N_HR; j += 1024) {
        float p = __expf(pi_hr[j] - maxpi) * invSum;
        z += __expf(p);
    }
    red[t] = z; __syncthreads();
    for (int off = 512; off > 0; off >>= 1) {
        if (t < off) red[t] += red[t + off];
        __syncthreads();
    }
    if (t == 0) s_logZ = logf(red[0]);
    __syncthreads();
    const float logZ = s_logZ;

    // 4) per-component params
    for (int j = t; j < N_HR; j += 1024) {
        float p      = __expf(pi_hr[j] - maxpi) * invSum;
        float logmix = p - logZ;
        float m0 = loc_hr[2 * j + 0] * (1.0f / 512.0f) + mu_hr[j];
        float m1 = loc_hr[2 * j + 1] * (1.0f / 512.0f) + mu_hr[N_HR + j];
        float s0 = sig_hr[j], s1 = sig_hr[N_HR + j];
        float a0 = 1.0f / (s0 * s0), a1 = 1.0f / (s1 * s1);
        float D  = -0.5f * (m0 * m0 * a0 + m1 * m1 * a1)
                   - logf(s0) - logf(s1) - LOG_2PI + logmix;
        float* rec = hrp + (size_t)j * 8;
        rec[0] = -0.5f * a0; rec[1] = m0 * a0;
        rec[2] = -0.5f * a1; rec[3] = m1 * a1;
        rec[4] = D; rec[5] = 0.f; rec[6] = 0.f; rec[7] = 0.f;
    }
}

// ---------------------------------------------------------------------------
// Prep LR: A-matrix features [t0^2, t0, t1^2, t1] per point.
// ---------------------------------------------------------------------------
__global__ void prep_lr_kernel(const float* __restrict__ mu_lr,
                               const float* __restrict__ loc_lr,
                               float* __restrict__ feat) {
    int i = blockIdx.x * blockDim.x + threadIdx.x;
    if (i < N_LR) {
        float t0 = loc_lr[2 * i + 0] * (1.0f / 512.0f) + mu_lr[i];
        float t1 = loc_lr[2 * i + 1] * (1.0f / 512.0f) + mu_lr[N_LR + i];
        float4 f; f.x = t0 * t0; f.y = t0; f.z = t1 * t1; f.w = t1;
        reinterpret_cast<float4*>(feat)[i] = f;
    }
}

// ---------------------------------------------------------------------------
// Main: rank-4 FP32 WMMA (16x16x4) + 1-exp streaming logsumexp.
// Work unit = (i-tile, j-slice): 2304 wave32s = 288 blocks x 8 waves.
// ---------------------------------------------------------------------------
__global__ __launch_bounds__(256)
void lse_main_kernel(const float* __restrict__ feat, const float* __restrict__ hrp,
                     float* __restrict__ partM, float* __restrict__ partS) {
    const int unit  = blockIdx.x * 8 + (threadIdx.x >> 5);
    const int itile = unit >> 4;            // 0..143
    const int slice = unit & (NSLICE - 1);  // 0..15
    const int lane  = threadIdx.x & 31;
    const int half  = lane >> 4;            // 0: lanes 0-15, 1: lanes 16-31
    const int ln    = lane & 15;

    // A-matrix 16x4 f32 (ISA layout): lanes 0-15 hold K=0,1 in v0,v1;
    // lanes 16-31 hold K=2,3. Row M = lane&15.
    const int M  = itile * 16 + ln;
    const int k0 = half * 2;
    v2f a;
    a.x = feat[M * 4 + k0];
    a.y = feat[M * 4 + k0 + 1];

    float m[8], s[8];
#pragma unroll
    for (int r = 0; r < 8; ++r) { m[r] = -3.0e38f; s[r] = 0.f; }

    const int j0 = slice * (JT_PER_SLICE * 16);
#pragma unroll 2
    for (int jt = 0; jt < JT_PER_SLICE; ++jt) {
        const int col    = j0 + jt * 16 + ln;
        const float* rec = hrp + (size_t)col * 8;

        // B-matrix 4x16 f32 (mirror of A layout): lanes 0-15 supply K=0,1,
        // lanes 16-31 supply K=2,3, column N = lane&15.
        const float2 bp = *reinterpret_cast<const float2*>(rec + half * 2);
        v2f b; b.x = bp.x; b.y = bp.y;

        // C = per-column constant D(j) broadcast over rows -> WMMA adds it free.
        const float D = rec[4];
        v8f c;
#pragma unroll
        for (int r = 0; r < 8; ++r) c[r] = D;

        v8f acc = __builtin_amdgcn_wmma_f32_16x16x4_f32(
            /*neg_a=*/false, a, /*neg_b=*/false, b,
            /*c_mod=*/(short)0, c, /*reuse_a=*/false, /*reuse_b=*/false);

#pragma unroll
        for (int r = 0; r < 8; ++r) lse_push(m[r], s[r], acc[r]);
    }

    // Combine the 16 per-column partials within each 16-lane half
    // (C/D layout: lanes 0-15 = rows 0-7, lanes 16-31 = rows 8-15).
#pragma unroll
    for (int mask = 8; mask >= 1; mask >>= 1) {
#pragma unroll
        for (int r = 0; r < 8; ++r) {
            float om = __shfl_xor(m[r], mask, 32);
            float os = __shfl_xor(s[r], mask, 32);
            lse_merge(m[r], s[r], om, os);
        }
    }

    if (ln == 0) {
#pragma unroll
        for (int r = 0; r < 8; ++r) {
            int row = itile * 16 + half * 8 + r;
            partM[row * NSLICE + slice] = m[r];
            partS[row * NSLICE + slice] = s[r];
        }
    }
}

// ---------------------------------------------------------------------------
// Finalize: merge slice partials -> log_probs[i]; loss = -sum(log_probs)/N_lr^2
// (mean(softmax(pi_lr)) == 1/N_lr since softmax sums to 1).
// ---------------------------------------------------------------------------
__global__ __launch_bounds__(256)
void finalize_kernel(const float* __restrict__ partM, const float* __restrict__ partS,
                     float* __restrict__ out) {
    __shared__ float red[256];
    const int t = threadIdx.x;
    float acc = 0.f;
    for (int i = t; i < N_LR; i += 256) {
        float m = -3.0e38f, s = 0.f;
#pragma unroll
        for (int k = 0; k < NSLICE; ++k) {
            lse_merge(m, s, partM[i * NSLICE + k], partS[i * NSLICE + k]);
        }
        acc += m + logf(s);      // log_probs[i]
    }
    red[t] = acc; __syncthreads();
    for (int off = 128; off > 0; off >>= 1) {
        if (t < off) red[t] += red[t + off];
        __syncthreads();
    }
    if (t == 0) out[0] = -red[0] / ((float)N_LR * (float)N_LR);
}

// ---------------------------------------------------------------------------
extern "C" void kernel_launch(void* const* d_in, const int* in_sizes, int n_in,
                              void* d_out, int out_size, void* d_ws, size_t ws_size,
                              hipStream_t stream) {
    const float* mu_lr  = (const float*)d_in[0];
    /* d_in[1] sigma_grid_lr: unused by the loss */
    /* d_in[2] pi_grid_lr: only enters as mean(softmax)=1/N_lr (softmax sums to 1) */
    const float* loc_lr = (const float*)d_in[3];
    const float* mu_hr  = (const float*)d_in[4];
    const float* sig_hr = (const float*)d_in[5];
    const float* pi_hr  = (const float*)d_in[6];
    const float* loc_hr = (const float*)d_in[7];
    /* d_in[8] topk = -1: default path */

    float* ws    = (float*)d_ws;
    float* hrp   = ws;                         // N_HR * 8 floats (512 KB)
    float* feat  = hrp + (size_t)N_HR * 8;     // N_LR * 4 floats
    float* partM = feat + (size_t)N_LR * 4;    // N_LR * NSLICE
    float* partS = partM + (size_t)N_LR * NSLICE;

    prep_hr_kernel<<<1, 1024, 0, stream>>>(mu_hr, sig_hr, pi_hr, loc_hr, hrp);
    prep_lr_kernel<<<(N_LR + 255) / 256, 256, 0, stream>>>(mu_lr, loc_lr, feat);
    lse_main_kernel<<<(N_LR / 16) * NSLICE / 8, 256, 0, stream>>>(feat, hrp, partM, partS);
    finalize_kernel<<<1, 256, 0, stream>>>(partM, partS, (float*)d_out);
}